// YOLOXPoseHead_46411416600624
// MI455X (gfx1250) — compile-verified
//
#include <hip/hip_runtime.h>
#include <stdint.h>

#define BATCH 128
#define NPRI  8400
#define NKPT  17
#define KSEL  300
#define PADK  304      // 19 * 16
#define TILES 19
#define IOUT  0.65f
#define FEPS  1e-7f

typedef __attribute__((ext_vector_type(2))) float v2f;
typedef __attribute__((ext_vector_type(8))) float v8f;

__device__ __forceinline__ float sigm(float x) { return 1.0f / (1.0f + expf(-x)); }

// MlvlPointGenerator priors: levels 80x80/s8, 40x40/s16, 20x20/s32, offset 0.5
__device__ __forceinline__ void prior_of(int n, float& px, float& py, float& st) {
  if (n < 6400)      { int y = n / 80;            int x = n - y * 80;          st = 8.f;  px = (x + 0.5f) * 8.f;  py = (y + 0.5f) * 8.f; }
  else if (n < 8000) { int m = n - 6400; int y = m / 40; int x = m - y * 40;   st = 16.f; px = (x + 0.5f) * 16.f; py = (y + 0.5f) * 16.f; }
  else               { int m = n - 8000; int y = m / 20; int x = m - y * 20;   st = 32.f; px = (x + 0.5f) * 32.f; py = (y + 0.5f) * 32.f; }
}

// order-preserving float -> uint32 (larger key == larger float)
__device__ __forceinline__ unsigned fkey(float f) {
  unsigned u = __float_as_uint(f);
  return (u & 0x80000000u) ? ~u : (u | 0x80000000u);
}

// ---------------- stage 1: fused score = sigmoid(cls) * sigmoid(obj) ----------------
__global__ void k_scores(const float* __restrict__ cls, const float* __restrict__ obj,
                         float* __restrict__ sc) {
  int i = blockIdx.x * blockDim.x + threadIdx.x;
  if (i < BATCH * NPRI) sc[i] = sigm(cls[i]) * sigm(obj[i]);
}

// ---------------- stage 2: per-batch exact top-300 (radix select + bitonic) ----------------
__global__ void __launch_bounds__(256) k_select(const float* __restrict__ sc,
                                                int* __restrict__ selIdx,
                                                float* __restrict__ selSc) {
  __shared__ unsigned hist[256];
  __shared__ unsigned s_pfx;
  __shared__ int s_rem, s_cnt;
  __shared__ unsigned long long cand[512];
  const int b = blockIdx.x, tid = threadIdx.x;
  const float* s = sc + (size_t)b * NPRI;

  // 4 passes of 8-bit MSB-first radix select for the 300th-largest key
  unsigned pfx = 0; int rem = KSEL;
  for (int pass = 0; pass < 4; ++pass) {
    int shift = 24 - pass * 8;
    __syncthreads();
    hist[tid] = 0;
    __syncthreads();
    for (int n = tid; n < NPRI; n += 256) {
      unsigned k = fkey(s[n]);
      bool match = true;
      if (pass > 0) { int hs = shift + 8; match = ((k >> hs) == (pfx >> hs)); }
      if (match) atomicAdd(&hist[(k >> shift) & 255u], 1u);
    }
    __syncthreads();
    if (tid == 0) {
      int r = rem;
      for (int bin = 255; bin >= 0; --bin) {
        int c = (int)hist[bin];
        if (c >= r) { s_pfx = pfx | ((unsigned)bin << shift); s_rem = r; break; }
        r -= c;
      }
    }
    __syncthreads();
    pfx = s_pfx; rem = s_rem;
  }

  // compact all keys >= threshold (count >= 300 by construction; cap 512)
  if (tid == 0) s_cnt = 0;
  __syncthreads();
  for (int n = tid; n < NPRI; n += 256) {
    unsigned k = fkey(s[n]);
    if (k >= pfx) {
      int p = atomicAdd(&s_cnt, 1);
      if (p < 512)  // composite: key desc major, index asc minor (matches top_k tie order)
        cand[p] = ((unsigned long long)k << 32) | (unsigned)(0xFFFFFFFFu - (unsigned)n);
    }
  }
  __syncthreads();
  int cnt = s_cnt; if (cnt > 512) cnt = 512;
  for (int i = tid + cnt; i < 512; i += 256) cand[i] = 0ull;
  __syncthreads();

  // bitonic sort 512 descending in LDS
  for (int k2 = 2; k2 <= 512; k2 <<= 1) {
    for (int j = k2 >> 1; j > 0; j >>= 1) {
      for (int i = tid; i < 512; i += 256) {
        int ixj = i ^ j;
        if (ixj > i) {
          unsigned long long a = cand[i], c = cand[ixj];
          bool descHalf = ((i & k2) == 0);
          if (descHalf ? (a < c) : (a > c)) { cand[i] = c; cand[ixj] = a; }
        }
      }
      __syncthreads();
    }
  }
  for (int i = tid; i < KSEL; i += 256) {
    unsigned long long w = cand[i];
    int idx = (int)(0xFFFFFFFFu - (unsigned)(w & 0xFFFFFFFFu));
    selIdx[b * KSEL + i] = idx;
    selSc[b * KSEL + i]  = s[idx];
  }
}

// ---------------- stage 3: decode boxes, WMMA-assisted IoU bitmatrix, serial NMS scan, emit ----------------
__global__ void __launch_bounds__(256) k_nms_out(const float* __restrict__ bbox,
                                                 const float* __restrict__ kofs,
                                                 const float* __restrict__ kvis,
                                                 const int* __restrict__ selIdx,
                                                 const float* __restrict__ selSc,
                                                 float* __restrict__ out) {
  __shared__ float4  sBox[PADK];
  __shared__ float   sArea[PADK];
  __shared__ unsigned sSup[PADK * 10];   // 304 rows x 320-bit suppression mask
  __shared__ unsigned sKeep[10];
  const int b = blockIdx.x, tid = threadIdx.x;
  const int lane = tid & 31, wid = tid >> 5;

  for (int i = tid; i < PADK * 10; i += 256) sSup[i] = 0u;

  // gather+decode the 300 selected boxes (pad 300..303 with zero boxes -> IoU 0)
  for (int i = tid; i < PADK; i += 256) {
    float4 bx = make_float4(0.f, 0.f, 0.f, 0.f); float ar = 0.f;
    if (i < KSEL) {
      int idx = selIdx[b * KSEL + i];
      float4 v = *(const float4*)(bbox + ((size_t)b * NPRI + idx) * 4);
      float px, py, st; prior_of(idx, px, py, st);
      float cx = v.x * st + px, cy = v.y * st + py;
      float w  = expf(v.z) * st, h = expf(v.w) * st;
      bx = make_float4(cx - w * 0.5f, cy - h * 0.5f, cx + w * 0.5f, cy + h * 0.5f);
      ar = fmaxf(bx.z - bx.x, 0.f) * fmaxf(bx.w - bx.y, 0.f);
    }
    sBox[i] = bx; sArea[i] = ar;
  }
  __syncthreads();

  // 19x19 tiles of 16x16; one tile per wave iteration. Area-sum matrix via
  // V_WMMA_F32_16X16X4_F32: D = [area_row | 1] x [1 ; area_col] = area_i + area_j.
  for (int t = wid; t < TILES * TILES; t += 8) {
    int ti = t / TILES, tj = t - ti * TILES;
    int rb = ti * 16, cb = tj * 16;
    int li = rb + (lane & 15), lj = cb + (lane & 15);
    v2f A, Bv;
    A.x  = (lane < 16) ? sArea[li] : 0.f;  // A: K=0 (lanes 0-15), K=2 (16-31)=0
    A.y  = (lane < 16) ? 1.0f      : 0.f;  //    K=1               K=3      =0
    Bv.x = (lane < 16) ? 1.0f      : 0.f;  // B: K=0 row = ones
    Bv.y = (lane < 16) ? sArea[lj] : 0.f;  //    K=1 row = area_col
    v8f C = {};
    C = __builtin_amdgcn_wmma_f32_16x16x4_f32(false, A, false, Bv, (short)0, C, false, false);

    int j = cb + (lane & 15);
    int mhi = (lane >> 4) * 8;
    float4 bj = sBox[j];
#pragma unroll
    for (int r = 0; r < 8; ++r) {
      int i = rb + mhi + r;                // C layout: lanes0-15 M=r, lanes16-31 M=r+8
      float4 bi = sBox[i];
      float ltx = fmaxf(bi.x, bj.x), lty = fmaxf(bi.y, bj.y);
      float rbx = fminf(bi.z, bj.z), rby = fminf(bi.w, bj.w);
      float inter = fmaxf(rbx - ltx, 0.f) * fmaxf(rby - lty, 0.f);
      float denom = (C[r] - inter) + FEPS;          // (area_i+area_j) - inter + eps
      bool p = (inter / denom > IOUT) && (j > i);
      unsigned m = __builtin_amdgcn_ballot_w32(p);
      unsigned lo = (m & 0xFFFFu) << ((tj & 1) * 16);
      unsigned hi = (m >> 16)     << ((tj & 1) * 16);
      if (lane == 0  && lo) atomicOr(&sSup[(rb + r) * 10 + (tj >> 1)], lo);
      if (lane == 16 && hi) atomicOr(&sSup[(rb + 8 + r) * 10 + (tj >> 1)], hi);
    }
  }
  __syncthreads();

  // serial suppression scan: single wave, 10 keep-words in registers, no barriers
  if (wid == 0) {
    unsigned kw = (lane < 10) ? 0xFFFFFFFFu : 0u;
    for (int i = 0; i < KSEL; ++i) {
      unsigned wrow = __shfl(kw, i >> 5, 32);
      if ((wrow >> (i & 31)) & 1u) {
        unsigned sv = (lane < 10) ? sSup[i * 10 + lane] : 0u;
        kw &= ~sv;
      }
    }
    if (lane < 10) sKeep[lane] = kw;
  }
  __syncthreads();

  // emit [300, 56]: box(4), score*keep(1), kpts(34), vis(17)
  for (int k = tid; k < KSEL; k += 256) {
    int idx = selIdx[b * KSEL + k];
    float sc = selSc[b * KSEL + k];
    float keep = ((sKeep[k >> 5] >> (k & 31)) & 1u) ? 1.0f : 0.0f;
    float px, py, st; prior_of(idx, px, py, st);
    float* o = out + ((size_t)b * KSEL + k) * 56;
    float4 bx = sBox[k];
    o[0] = bx.x; o[1] = bx.y; o[2] = bx.z; o[3] = bx.w;
    o[4] = sc * keep;
    const float* kb = kofs + ((size_t)b * NPRI + idx) * NKPT * 2;
    const float* vb = kvis + ((size_t)b * NPRI + idx) * NKPT;
#pragma unroll
    for (int q = 0; q < NKPT; ++q) {
      o[5 + 2 * q]     = kb[2 * q]     * st + px;
      o[5 + 2 * q + 1] = kb[2 * q + 1] * st + py;
    }
#pragma unroll
    for (int q = 0; q < NKPT; ++q) o[39 + q] = sigm(vb[q]);
  }
}

extern "C" void kernel_launch(void* const* d_in, const int* in_sizes, int n_in,
                              void* d_out, int out_size, void* d_ws, size_t ws_size,
                              hipStream_t stream) {
  (void)in_sizes; (void)n_in; (void)out_size; (void)ws_size;
  const float* cls  = (const float*)d_in[0];   // [B,N,1]
  const float* bbox = (const float*)d_in[1];   // [B,N,4]
  const float* obj  = (const float*)d_in[2];   // [B,N]
  const float* kofs = (const float*)d_in[3];   // [B,N,17,2]
  const float* kvis = (const float*)d_in[4];   // [B,N,17]
  float* out = (float*)d_out;                  // [B,300,56]

  float* sc     = (float*)d_ws;                                         // B*N floats
  int*   selIdx = (int*)((char*)d_ws + (size_t)BATCH * NPRI * 4);       // B*300 ints
  float* selSc  = (float*)((char*)selIdx + (size_t)BATCH * KSEL * 4);   // B*300 floats

  k_scores<<<(BATCH * NPRI + 255) / 256, 256, 0, stream>>>(cls, obj, sc);
  k_select<<<BATCH, 256, 0, stream>>>(sc, selIdx, selSc);
  k_nms_out<<<BATCH, 256, 0, stream>>>(bbox, kofs, kvis, selIdx, selSc, out);
}